// WarpingLayerFWFlow_67783173866031
// MI455X (gfx1250) — compile-verified
//
#include <hip/hip_runtime.h>
#include <stdint.h>

// Forward-warp (z-buffered scatter) for N=8, C=3, H=720, W=1280 fp32.
//
// Pipeline (all on `stream`, graph-capture safe):
//   1) ws_zero_kernel     : zero 59 MB of u64 keys in d_ws
//   2) fwdwarp_scatter    : per source pixel, key = (bits(|flow|^2)<<32)|(p+1),
//                           global_atomic_max_u64 into ws[target]   (winner =
//                           max |flow|^2, ties -> max source index; matches the
//                           reference's stable-argsort-rank semantics exactly)
//   3) fwdwarp_gather     : per target pixel, decode winning source, gather 3
//                           channels, holes -> 0
//
// Memory-bound problem (~0.4 GB traffic, ~17us floor at 23.3 TB/s). No matmul
// structure -> WMMA inapplicable. CDNA5-specific tuning: u64 L2 atomics, B128
// vector accesses (4 px/thread, wave32-friendly 256-thread blocks = 8 waves),
// and TH_NT temporal hints on the streaming traffic (flow in, keys in gather,
// out stores) so the reusable set (image planes + ws keys, ~148 MB) stays
// resident in the 192 MB L2.

typedef float               v4f   __attribute__((ext_vector_type(4)));
typedef unsigned int        v4u   __attribute__((ext_vector_type(4)));
typedef unsigned long long  v2u64 __attribute__((ext_vector_type(2)));

#define IMG_W 1280
#define IMG_H 720
#define IMG_HW (IMG_W * IMG_H)      // 921600
#define NBATCH 8
#define NCHAN 3
#define QUADS_PER_IMG (IMG_HW / 4)  // 230400

// ---------------------------------------------------------------------------
// 1) Zero the key buffer: each thread clears 32 bytes (two b128 stores).
//    Regular (RT) stores: these lines are about to be hammered by L2 atomics,
//    we WANT them resident.
__global__ __launch_bounds__(256) void ws_zero_kernel(v4u* __restrict__ ws) {
  size_t t = (size_t)blockIdx.x * 256 + threadIdx.x;
  v4u z = (v4u)0u;
  ws[2 * t + 0] = z;
  ws[2 * t + 1] = z;
}

// ---------------------------------------------------------------------------
// 2) Scatter: one thread = 4 consecutive source pixels of one batch image.
__global__ __launch_bounds__(256) void fwdwarp_scatter_kernel(
    const float* __restrict__ flow, unsigned long long* __restrict__ ws) {
  int t  = blockIdx.x * 256 + threadIdx.x;
  int n  = t / QUADS_PER_IMG;
  int q  = t - n * QUADS_PER_IMG;
  int p0 = q * 4;                  // linear pixel index of first of 4
  int my  = p0 / IMG_W;            // whole quad is in one row (W % 4 == 0)
  int mx0 = p0 - my * IMG_W;

  const float* fb = flow + (size_t)n * 2 * IMG_HW;
  // flow is read exactly once -> non-temporal so it doesn't evict L2 residents
  v4f fx4 = __builtin_nontemporal_load((const v4f*)(fb + p0));
  v4f fy4 = __builtin_nontemporal_load((const v4f*)(fb + IMG_HW + p0));

  unsigned long long* wsb = ws + (size_t)n * IMG_HW;
#pragma unroll
  for (int k = 0; k < 4; ++k) {
    float fx = fx4[k];
    float fy = fy4[k];
    // trunc-toward-zero int conversion == jnp .astype(int32)
    int wx = mx0 + k + (int)fx;
    int wy = my      + (int)fy;
    wx = min(max(wx, 0), IMG_W - 1);
    wy = min(max(wy, 0), IMG_H - 1);
    // mul,mul,add with no FMA contraction -> bit-match the reference's v
    float v = __fadd_rn(__fmul_rn(fx, fx), __fmul_rn(fy, fy));
    // v >= 0 so its bit pattern is order-preserving as u32.
    // Low word p+1: larger source index wins ties; 0 == hole sentinel.
    unsigned long long key =
        ((unsigned long long)__float_as_uint(v) << 32) |
        (unsigned long long)(unsigned)(p0 + k + 1);
    atomicMax(wsb + (wy * IMG_W + wx), key);   // global_atomic_max_u64 @ L2
  }
}

// ---------------------------------------------------------------------------
// 3) Gather: one thread = 4 consecutive target pixels; decode winner, fetch
//    all 3 channels, write holes as 0. Branch-free (clamped gather + select)
//    to keep the wave32 EXEC mask full through the memory ops.
__global__ __launch_bounds__(256) void fwdwarp_gather_kernel(
    const float* __restrict__ image,
    const unsigned long long* __restrict__ ws,
    float* __restrict__ out) {
  int t  = blockIdx.x * 256 + threadIdx.x;
  int n  = t / QUADS_PER_IMG;
  int p0 = (t - n * QUADS_PER_IMG) * 4;

  const unsigned long long* wsb = ws + (size_t)n * IMG_HW;
  // keys are read exactly once after the scatter -> non-temporal
  v2u64 k01 = __builtin_nontemporal_load((const v2u64*)(wsb + p0));
  v2u64 k23 = __builtin_nontemporal_load((const v2u64*)(wsb + p0 + 2));
  unsigned long long keys[4] = {k01.x, k01.y, k23.x, k23.y};

  const float* img = image + (size_t)n * NCHAN * IMG_HW;
  v4f o0, o1, o2;
#pragma unroll
  for (int k = 0; k < 4; ++k) {
    unsigned long long key = keys[k];
    int src = (int)(unsigned)(key & 0xFFFFFFFFull) - 1;  // -1 on hole
    src = max(src, 0);                                    // safe dummy gather
    // default (RT) loads: winners repeat across targets; keep image in L2
    float a = img[src];
    float b = img[IMG_HW + src];
    float c = img[2 * IMG_HW + src];
    bool cov = (key != 0ull);
    o0[k] = cov ? a : 0.0f;
    o1[k] = cov ? b : 0.0f;
    o2[k] = cov ? c : 0.0f;
  }

  float* ob = out + (size_t)n * NCHAN * IMG_HW + p0;
  // output is write-once, never re-read by us -> non-temporal b128 stores
  __builtin_nontemporal_store(o0, (v4f*)(ob));
  __builtin_nontemporal_store(o1, (v4f*)(ob + IMG_HW));
  __builtin_nontemporal_store(o2, (v4f*)(ob + 2 * IMG_HW));
}

// ---------------------------------------------------------------------------
extern "C" void kernel_launch(void* const* d_in, const int* in_sizes, int n_in,
                              void* d_out, int out_size, void* d_ws, size_t ws_size,
                              hipStream_t stream) {
  const float* image = (const float*)d_in[0];  // (8,3,720,1280) f32
  const float* flow  = (const float*)d_in[1];  // (8,2,720,1280) f32
  float* out = (float*)d_out;                  // (8,3,720,1280) f32
  unsigned long long* ws = (unsigned long long*)d_ws;  // 8*921600 u64 = ~59 MB

  const int total_quads = NBATCH * QUADS_PER_IMG;  // 1,843,200
  const int blocks      = total_quads / 256;       // 7,200 (exact)

  // blocks*256 threads * 32 B each == NBATCH*IMG_HW*8 B exactly
  ws_zero_kernel<<<blocks, 256, 0, stream>>>((v4u*)d_ws);
  fwdwarp_scatter_kernel<<<blocks, 256, 0, stream>>>(flow, ws);
  fwdwarp_gather_kernel<<<blocks, 256, 0, stream>>>(image, ws, out);

  (void)in_sizes; (void)n_in; (void)out_size; (void)ws_size;
}